// MCCLoss_77807627535079
// MI455X (gfx1250) — compile-verified
//
#include <hip/hip_runtime.h>
#include <hip/hip_bf16.h>
#include <math.h>

// MCC loss for MI455X (gfx1250, wave32).
// Pipeline:
//   Kernel 1 (HBM-bound streaming pass, reads each input exactly once):
//     per-wave row normalization + column accumulation -> per-block partial colsums in d_ws
//   Kernel 2 (tiny): fold block partials -> s[512] per feature, then
//     dot(s,s) = trace(A^T A) via 8x accumulating V_WMMA_F32_16X16X4_F32, final scalar.

typedef __attribute__((ext_vector_type(2))) float v2f;
typedef __attribute__((ext_vector_type(8))) float v8f;

#define N_ROWS 16384
#define DCOLS  512
#define NBLK   256
#define TPB    256
#define WAVES_PER_BLK (TPB / 32)
#define TOTAL_WAVES   (NBLK * WAVES_PER_BLK)

__device__ __forceinline__ float wave_reduce_sum(float v) {
    v += __shfl_xor(v, 1, 32);
    v += __shfl_xor(v, 2, 32);
    v += __shfl_xor(v, 4, 32);
    v += __shfl_xor(v, 8, 32);
    v += __shfl_xor(v, 16, 32);
    return v;
}

// ---------------------------------------------------------------------------
// Kernel 1: one wave owns a full 512-col row (16 f32/lane via 4x b128 loads).
// Normalize in-register, accumulate columns. Per-block partials -> workspace.
// ---------------------------------------------------------------------------
__global__ __launch_bounds__(TPB) void mcc_partial_kernel(
    const float* __restrict__ sk, const float* __restrict__ ph,
    float* __restrict__ psk, float* __restrict__ pph)
{
    __shared__ float lds_sk[WAVES_PER_BLK][DCOLS];
    __shared__ float lds_ph[WAVES_PER_BLK][DCOLS];

    const int lane = threadIdx.x & 31;
    const int wv   = threadIdx.x >> 5;
    const int gw   = blockIdx.x * WAVES_PER_BLK + wv;

    float acc_sk[16], acc_ph[16];
#pragma unroll
    for (int i = 0; i < 16; ++i) { acc_sk[i] = 0.0f; acc_ph[i] = 0.0f; }

    for (int r = gw; r < N_ROWS; r += TOTAL_WAVES) {
        // ---- sketch feature row ----
        {
            const float4* row = (const float4*)(sk + (size_t)r * DCOLS);
            float4 v[4];
#pragma unroll
            for (int k = 0; k < 4; ++k) v[k] = row[k * 32 + lane];
            float ss = 0.0f;
#pragma unroll
            for (int k = 0; k < 4; ++k)
                ss += v[k].x * v[k].x + v[k].y * v[k].y +
                      v[k].z * v[k].z + v[k].w * v[k].w;
            ss = wave_reduce_sum(ss);
            const float inv = 1.0f / fmaxf(sqrtf(ss), 1e-12f);
#pragma unroll
            for (int k = 0; k < 4; ++k) {
                acc_sk[k * 4 + 0] += v[k].x * inv;
                acc_sk[k * 4 + 1] += v[k].y * inv;
                acc_sk[k * 4 + 2] += v[k].z * inv;
                acc_sk[k * 4 + 3] += v[k].w * inv;
            }
        }
        // ---- photo feature row ----
        {
            const float4* row = (const float4*)(ph + (size_t)r * DCOLS);
            float4 v[4];
#pragma unroll
            for (int k = 0; k < 4; ++k) v[k] = row[k * 32 + lane];
            float ss = 0.0f;
#pragma unroll
            for (int k = 0; k < 4; ++k)
                ss += v[k].x * v[k].x + v[k].y * v[k].y +
                      v[k].z * v[k].z + v[k].w * v[k].w;
            ss = wave_reduce_sum(ss);
            const float inv = 1.0f / fmaxf(sqrtf(ss), 1e-12f);
#pragma unroll
            for (int k = 0; k < 4; ++k) {
                acc_ph[k * 4 + 0] += v[k].x * inv;
                acc_ph[k * 4 + 1] += v[k].y * inv;
                acc_ph[k * 4 + 2] += v[k].z * inv;
                acc_ph[k * 4 + 3] += v[k].w * inv;
            }
        }
    }

    // Column owned by (lane, k, j) is k*128 + lane*4 + j.
#pragma unroll
    for (int k = 0; k < 4; ++k) {
#pragma unroll
        for (int j = 0; j < 4; ++j) {
            const int c = k * 128 + lane * 4 + j;
            lds_sk[wv][c] = acc_sk[k * 4 + j];
            lds_ph[wv][c] = acc_ph[k * 4 + j];
        }
    }
    __syncthreads();

    // Fold the 8 waves' partials; write this block's partial colsums (no atomics).
    for (int c = threadIdx.x; c < DCOLS; c += TPB) {
        float s0 = 0.0f, s1 = 0.0f;
#pragma unroll
        for (int w = 0; w < WAVES_PER_BLK; ++w) {
            s0 += lds_sk[w][c];
            s1 += lds_ph[w][c];
        }
        psk[(size_t)blockIdx.x * DCOLS + c] = s0;
        pph[(size_t)blockIdx.x * DCOLS + c] = s1;
    }
}

// ---------------------------------------------------------------------------
// Kernel 2: fold NBLK partials -> s[512]; dot(s,s) = trace(A^T A) via WMMA.
// A is 16x4 f32 per ISA layout: lanes 0-15 -> M=lane, K=0,1 in V0,V1;
// lanes 16-31 -> M=lane-16, K=2,3. Using the same registers for A and B,
// C[m,n] = sum_k s[b+m*4+k]*s[b+n*4+k]; accumulating over 8 chunks of 64,
// trace(C) = dot(s,s). Diagonal: lane l holds C[l,l] in c[l] (l<8) and
// C[l-16+?]: lanes 24-31 hold C[m,m], m=8..15, in c[lane-24].
// ---------------------------------------------------------------------------
__global__ __launch_bounds__(512) void mcc_final_kernel(
    const float* __restrict__ psk, const float* __restrict__ pph,
    float* __restrict__ out)
{
    __shared__ float ls[2][DCOLS];

    const int t = threadIdx.x;
    if (t < DCOLS) {
        float s0 = 0.0f, s1 = 0.0f;
        for (int b = 0; b < NBLK; ++b) {       // coalesced: 512 threads x row b
            s0 += psk[(size_t)b * DCOLS + t];
            s1 += pph[(size_t)b * DCOLS + t];
        }
        ls[0][t] = s0;
        ls[1][t] = s1;
    }
    __syncthreads();

    if (t < 32) {                               // whole wave 0: EXEC all-1s
        const int lane = t;
        const int m    = lane & 15;
        const int kb   = (lane >> 4) * 2;

        float dots[2];
#pragma unroll
        for (int f = 0; f < 2; ++f) {
            v8f c = {};
#pragma unroll
            for (int q = 0; q < 8; ++q) {       // 8 chunks of 64 elements
                v2f a;
                a.x = ls[f][q * 64 + m * 4 + kb + 0];
                a.y = ls[f][q * 64 + m * 4 + kb + 1];
                c = __builtin_amdgcn_wmma_f32_16x16x4_f32(
                        /*neg_a=*/false, a, /*neg_b=*/false, a,
                        /*c_mod=*/(short)0, c,
                        /*reuse_a=*/false, /*reuse_b=*/false);
            }
            // Extract trace: lanes 0-7 -> c[lane] (M=N=0..7),
            //                lanes 24-31 -> c[lane-24] (M=N=8..15).
            float tr = 0.0f;
            if (lane < 8)        tr = c[lane];
            else if (lane >= 24) tr = c[lane - 24];
            tr = wave_reduce_sum(tr);
            dots[f] = tr;
        }

        if (lane == 0) {
            const float inv_n2 = 1.0f / ((float)N_ROWS * (float)N_ROWS);
            const float loss_sk = fabsf(dots[0] * inv_n2 - 0.1f);
            const float loss_ph = fabsf(dots[1] * inv_n2 - 0.1f);
            out[0] = loss_sk + loss_ph;         // lambdas are all 1.0
        }
    }
}

extern "C" void kernel_launch(void* const* d_in, const int* in_sizes, int n_in,
                              void* d_out, int out_size, void* d_ws, size_t ws_size,
                              hipStream_t stream) {
    const float* sk  = (const float*)d_in[0];
    const float* ph  = (const float*)d_in[1];
    // d_in[2] (neg_feat) is normalized in the reference but unused in the loss:
    // never touch it -> saves 33.5 MB of HBM traffic.
    float* out = (float*)d_out;

    float* psk = (float*)d_ws;                  // [NBLK][512]
    float* pph = psk + (size_t)NBLK * DCOLS;    // [NBLK][512]  (1 MB total)

    mcc_partial_kernel<<<NBLK, TPB, 0, stream>>>(sk, ph, psk, pph);
    mcc_final_kernel<<<1, 512, 0, stream>>>(psk, pph, out);
}